// GNN_base_77670188581209
// MI455X (gfx1250) — compile-verified
//
#include <hip/hip_runtime.h>
#include <hip/hip_bf16.h>

typedef __attribute__((ext_vector_type(2))) float v2f;
typedef __attribute__((ext_vector_type(8))) float v8f;

#define BATCH   16
#define SEQ     2048
#define FEAT    256
#define NVOCAB  8000
#define PAD     8           // P
#define NBR     16          // 2*P neighbors
#define CHUNK   32          // positions per block
#define NCHUNK  (SEQ / CHUNK)   // 64 chunks
#define WIN     (CHUNK + 2 * PAD) // 48 window rows

// ---------------------------------------------------------------------------
// Kernel 1: per-(batch, chunk) partial sums.
// Stage the 48-row embedding window + edge/node weights in LDS once, then each
// thread owns one feature column and accumulates over the 32 positions.
// ---------------------------------------------------------------------------
__global__ __launch_bounds__(256) void gnn_partial_kernel(
    const int* __restrict__ ids,        // (BATCH, SEQ)
    const float* __restrict__ node_emb, // (NVOCAB, FEAT), row 0 is zeros
    const float* __restrict__ edge_w,   // (NVOCAB*NVOCAB+1, 1)
    const float* __restrict__ node_w,   // (NVOCAB, 1)
    float* __restrict__ partial)        // (BATCH, NCHUNK, FEAT)
{
    __shared__ float s_emb[WIN * FEAT];   // 49152 B: staged embedding window
    __shared__ float s_ew[CHUNK * NBR];   //  2048 B: edge weights per (pos,nbr)
    __shared__ float s_nw[CHUNK];         // node weight per position
    __shared__ int   s_id[WIN];           // ids in window (0 when out of range)

    const int b     = blockIdx.y;
    const int chunk = blockIdx.x;
    const int l0    = chunk * CHUNK;
    const int tid   = threadIdx.x;

    // --- window ids (position l0-8 .. l0+39); out-of-sequence -> id 0 (pad)
    if (tid < WIN) {
        const int lp = l0 - PAD + tid;
        s_id[tid] = (lp >= 0 && lp < SEQ) ? ids[b * SEQ + lp] : 0;
    }
    __syncthreads();

    // --- stage embedding rows: coalesced 1KB-row loads, one row per iter.
    // node_emb row 0 is all-zero, so pad ids naturally contribute zeros.
    for (int r = 0; r < WIN; ++r) {
        s_emb[r * FEAT + tid] = node_emb[s_id[r] * FEAT + tid];
    }

    // --- node weights for the 32 center positions
    if (tid < CHUNK) {
        s_nw[tid] = node_w[s_id[tid + PAD]];
    }

    // --- edge weights: 32*16 = 512 gathers into the 256MB table, done once
    for (int e = tid; e < CHUNK * NBR; e += 256) {
        const int pos = e >> 4;       // 0..31
        const int k   = e & 15;       // 0..15
        const int w   = pos + k + ((k >= 8) ? 1 : 0);  // neighbor window idx
        const int nx  = s_id[w];
        const int idc = s_id[pos + PAD];
        // nx==0 (pad or token 0) -> index 0, matching jnp.where(NX==0, 0, EW).
        // idc*8000+nx <= 64,007,999 fits in int32.
        const int eidx = (nx == 0) ? 0 : (idc * NVOCAB + nx);
        s_ew[e] = edge_w[eidx];
    }
    __syncthreads();

    // --- compute: thread `tid` owns feature column f = tid.
    // LDS reads are bank-conflict-free (stride FEAT = 256 == 0 mod 64 banks).
    float acc = 0.0f;
    const int f = tid;
    for (int pos = 0; pos < CHUNK; ++pos) {
        const float nn = s_nw[pos];
        const float rn = s_emb[(pos + PAD) * FEAT + f];
        const float* ewp = &s_ew[pos * NBR];
        // neighbor k<8 -> window pos+k (offsets -8..-1); k>=8 -> pos+k+1 (+1..+8)
        float m = s_emb[pos * FEAT + f] * ewp[0];
#pragma unroll
        for (int k = 1; k < NBR; ++k) {
            const int w = pos + k + ((k >= 8) ? 1 : 0);
            m = fmaxf(m, s_emb[w * FEAT + f] * ewp[k]);
        }
        acc += (1.0f - nn) * m + nn * rn;
    }
    partial[(b * NCHUNK + chunk) * FEAT + f] = acc;
}

// ---------------------------------------------------------------------------
// Kernel 2: reduce partials over the 64 chunks with chained
// V_WMMA_F32_16X16X4_F32. A = all-ones 16x4, B[k][n] = partial chunk values,
// so D[m][n] accumulates sum_k B[k][n] exactly (f32). One wave per
// (batch, 16-feature group); 16 chained WMMAs cover 64 chunks.
// ---------------------------------------------------------------------------
__global__ __launch_bounds__(32) void gnn_reduce_wmma_kernel(
    const float* __restrict__ partial,  // (BATCH, NCHUNK, FEAT)
    float* __restrict__ out)            // (BATCH, FEAT)
{
    const int fg   = blockIdx.x;        // feature group 0..15
    const int b    = blockIdx.y;        // batch 0..15
    const int lane = threadIdx.x;       // 0..31
    const int n    = lane & 15;         // N (feature within group)
    const int hi   = lane >> 4;         // lane half selects K offset

    const float* pb = partial + b * (NCHUNK * FEAT) + fg * 16 + n;

    v2f a;                              // all-ones A (16x4): K labels irrelevant
    a.x = 1.0f;
    a.y = 1.0f;
    v8f c = {};

#pragma unroll
    for (int c0 = 0; c0 < NCHUNK; c0 += 4) {
        v2f bm;                         // B (4x16): 4 chunks x 16 features
        bm.x = pb[(c0 + 0 + hi * 2) * FEAT];
        bm.y = pb[(c0 + 1 + hi * 2) * FEAT];
        // (neg_a, A, neg_b, B, c_mod, C, reuse_a, reuse_b)
        c = __builtin_amdgcn_wmma_f32_16x16x4_f32(
            false, a, false, bm, (short)0, c, false, false);
    }

    // Every row of D holds the same sums; lane 0..15 write row M=0.
    if (lane < 16) {
        out[b * FEAT + fg * 16 + lane] = c[0];
    }
}

// ---------------------------------------------------------------------------
extern "C" void kernel_launch(void* const* d_in, const int* in_sizes, int n_in,
                              void* d_out, int out_size, void* d_ws, size_t ws_size,
                              hipStream_t stream) {
    const int*   ids      = (const int*)d_in[0];
    const float* node_emb = (const float*)d_in[1];
    const float* edge_w   = (const float*)d_in[2];
    const float* node_w   = (const float*)d_in[3];
    float*       out      = (float*)d_out;
    float*       partial  = (float*)d_ws;   // BATCH*NCHUNK*FEAT f32 = 1 MB

    dim3 grid1(NCHUNK, BATCH);
    gnn_partial_kernel<<<grid1, 256, 0, stream>>>(ids, node_emb, edge_w,
                                                  node_w, partial);

    dim3 grid2(FEAT / 16, BATCH);
    gnn_reduce_wmma_kernel<<<grid2, 32, 0, stream>>>(partial, out);
}